// FAE_SAGEConv_5231270167342
// MI455X (gfx1250) — compile-verified
//
#include <hip/hip_runtime.h>

typedef __attribute__((ext_vector_type(2))) float v2f;
typedef __attribute__((ext_vector_type(8))) float v8f;

#define N_NODES 50000
#define N_EDGES 1600000
#define K_DIM   64

// ---------------- utility: zero fill (grid-stride) ----------------
__global__ void zero_f32(float* __restrict__ p, int n) {
    int i = blockIdx.x * blockDim.x + threadIdx.x;
    int stride = gridDim.x * blockDim.x;
    for (; i < n; i += stride) p[i] = 0.0f;
}

// ---------------- degree count: cnt[dst] += 1 ----------------
__global__ void degree_count(const int* __restrict__ ei, float* __restrict__ cnt, int E) {
    int e = blockIdx.x * blockDim.x + threadIdx.x;
    if (e < E) atomicAdd(&cnt[ei[E + e]], 1.0f);
}

// ---------------- dual WMMA GEMM: Y = X @ Wl^T ; Z = X @ Wr^T + bl ----------------
// X: [M, 64] row-major. Wl/Wr: [NT*16, 64] row-major (W^T[k][n] = W[n*64+k]).
// Weights staged in LDS once per block; one wave per 16-row tile; NT column
// tiles of 16; K=64 in 16 steps of 4 via V_WMMA_F32_16X16X4_F32.
// A fragments for all K preloaded into registers; inner loop = ds_load + wmma.
template<int NT>
__global__ void gemm_dual_wmma(const float* __restrict__ X,
                               const float* __restrict__ Wl,
                               const float* __restrict__ Wr,
                               const float* __restrict__ bl,
                               float* __restrict__ Y,
                               float* __restrict__ Z,
                               int M) {
    const int NOUT = NT * 16;
    __shared__ float sWl[NT * 16 * K_DIM];
    __shared__ float sWr[NT * 16 * K_DIM];

    // cooperative stage of both weight matrices (NT*16*64 floats each)
    for (int i = threadIdx.x; i < NT * 16 * K_DIM / 4; i += blockDim.x) {
        ((float4*)sWl)[i] = ((const float4*)Wl)[i];
        ((float4*)sWr)[i] = ((const float4*)Wr)[i];
    }
    __syncthreads();

    const int lane = threadIdx.x & 31;
    const int wave = blockIdx.x * (blockDim.x >> 5) + (threadIdx.x >> 5);
    const int mt = wave;                       // 16-row tile index (wave-uniform)
    if (mt * 16 < M) {                         // uniform guard keeps EXEC full
        const int r = lane & 15;               // M (A) / N (B) index
        const int h = lane >> 4;               // lane-half selects K pair

        // Preload all 16 A fragments: lane reads X[mt*16+r][4*kk + 2h + {0,1}]
        const float* xrow = X + (size_t)(mt * 16 + r) * K_DIM;
        v2f a[16];
#pragma unroll
        for (int kk = 0; kk < 16; ++kk) {
            a[kk].x = xrow[4 * kk + 2 * h + 0];
            a[kk].y = xrow[4 * kk + 2 * h + 1];
        }

        v8f accY[NT], accZ[NT];
#pragma unroll
        for (int t = 0; t < NT; ++t) {
            accY[t] = (v8f){0.f,0.f,0.f,0.f,0.f,0.f,0.f,0.f};
            accZ[t] = (v8f){0.f,0.f,0.f,0.f,0.f,0.f,0.f,0.f};
        }

#pragma unroll
        for (int kk = 0; kk < 16; ++kk) {
            const int k = 4 * kk;
            // B fragment: B[k'][n] = W[n*64+k'], lane carries n = t*16 + r,
            // regs/half carry k' = k + 2h + {0,1} -> contiguous float2 in LDS row.
            v2f bL[NT], bR[NT];
#pragma unroll
            for (int t = 0; t < NT; ++t) {
                const float* lp = &sWl[(t * 16 + r) * K_DIM + k + 2 * h];
                bL[t].x = lp[0]; bL[t].y = lp[1];
                const float* rp = &sWr[(t * 16 + r) * K_DIM + k + 2 * h];
                bR[t].x = rp[0]; bR[t].y = rp[1];
            }
#pragma unroll
            for (int t = 0; t < NT; ++t) {
                accY[t] = __builtin_amdgcn_wmma_f32_16x16x4_f32(
                    false, a[kk], false, bL[t], (short)0, accY[t], false, false);
                accZ[t] = __builtin_amdgcn_wmma_f32_16x16x4_f32(
                    false, a[kk], false, bR[t], (short)0, accZ[t], false, false);
            }
        }

        // D layout: VGPR i -> row M = i + 8*h, col N = lane&15
#pragma unroll
        for (int t = 0; t < NT; ++t) {
            float bias = bl[t * 16 + r];
#pragma unroll
            for (int i = 0; i < 8; ++i) {
                int row = mt * 16 + i + 8 * h;
                Y[(size_t)row * NOUT + t * 16 + r] = accY[t][i];
                Z[(size_t)row * NOUT + t * 16 + r] = accZ[t][i] + bias;
            }
        }
    }
}

// ---------------- edge scatter: acc[dst] += Y[src], CH channels ----------------
// CH/4 threads per edge, float4 gather (contiguous 256B/128B per edge).
template<int CH>
__global__ void scatter_add(const float* __restrict__ Y,
                            const int* __restrict__ ei,
                            float* __restrict__ acc, int E) {
    const int G = CH / 4;
    int tid = blockIdx.x * blockDim.x + threadIdx.x;
    if (tid >= E * G) return;
    int e  = tid / G;
    int c4 = tid - e * G;
    int src = ei[e];
    int dst = ei[E + e];
    const float4 v = *(const float4*)&Y[(size_t)src * CH + c4 * 4];
    float* a = &acc[(size_t)dst * CH + c4 * 4];
    atomicAdd(a + 0, v.x);
    atomicAdd(a + 1, v.y);
    atomicAdd(a + 2, v.z);
    atomicAdd(a + 3, v.w);
}

// ---------------- combine: acc = relu(acc / max(cnt,1) + Z) (in place) ----------------
template<int CH>
__global__ void combine_relu(float* __restrict__ acc,
                             const float* __restrict__ Z,
                             const float* __restrict__ cnt, int n) {
    int i = blockIdx.x * blockDim.x + threadIdx.x;
    if (i >= n * CH) return;
    int node = i / CH;
    float inv = 1.0f / fmaxf(cnt[node], 1.0f);
    float v = acc[i] * inv + Z[i];
    acc[i] = fmaxf(v, 0.0f);
}

// ---------------- final linear: out = H @ Wlin^T + blin ----------------
__global__ void final_linear(const float* __restrict__ H,
                             const float* __restrict__ Wlin,
                             const float* __restrict__ blin,
                             float* __restrict__ out, int n) {
    int i = blockIdx.x * blockDim.x + threadIdx.x;
    if (i >= n) return;
    const float4* hp = (const float4*)&H[(size_t)i * 32];
    const float4* wp = (const float4*)Wlin;
    float s = 0.0f;
#pragma unroll
    for (int j = 0; j < 8; ++j) {
        float4 hv = hp[j], wv = wp[j];
        s += hv.x * wv.x + hv.y * wv.y + hv.z * wv.z + hv.w * wv.w;
    }
    out[i] = s + blin[0];
}

extern "C" void kernel_launch(void* const* d_in, const int* in_sizes, int n_in,
                              void* d_out, int out_size, void* d_ws, size_t ws_size,
                              hipStream_t stream) {
    const float* x    = (const float*)d_in[0];
    const int*   ei   = (const int*)  d_in[1];   // [2, E] : src row then dst row
    const float* W1l  = (const float*)d_in[2];
    const float* b1l  = (const float*)d_in[3];
    const float* W1r  = (const float*)d_in[4];
    const float* W2l  = (const float*)d_in[5];
    const float* b2l  = (const float*)d_in[6];
    const float* W2r  = (const float*)d_in[7];
    const float* Wlin = (const float*)d_in[8];
    const float* blin = (const float*)d_in[9];
    float* out = (float*)d_out;

    // workspace layout (floats):  cnt[N] | y1[N*64] | z1[N*64] | acc1[N*64]
    // reuse: h1 <- acc1 (in place); y2,z2 <- y1 region; acc2/h2 <- z1 region
    float* ws   = (float*)d_ws;
    float* cnt  = ws;
    float* y1   = cnt + N_NODES;
    float* z1   = y1 + (size_t)N_NODES * 64;
    float* acc1 = z1 + (size_t)N_NODES * 64;
    float* h1   = acc1;
    float* y2   = y1;
    float* z2   = y1 + (size_t)N_NODES * 32;
    float* acc2 = z1;                           // reused AFTER combine_relu<64>
    float* h2   = acc2;

    const int tb = 256;
    const int mtiles = N_NODES / 16;            // 3125 (exact)
    const int gemm_blocks = (mtiles + 7) / 8;   // 8 waves/block

    zero_f32<<<1024, tb, 0, stream>>>(cnt, N_NODES);
    zero_f32<<<4096, tb, 0, stream>>>(acc1, N_NODES * 64);
    degree_count<<<(N_EDGES + tb - 1) / tb, tb, 0, stream>>>(ei, cnt, N_EDGES);

    // ---- layer 1 (64 -> 64): transform first, then scatter ----
    gemm_dual_wmma<4><<<gemm_blocks, 256, 0, stream>>>(x, W1l, W1r, b1l, y1, z1, N_NODES);
    scatter_add<64><<<(N_EDGES * 16 + tb - 1) / tb, tb, 0, stream>>>(y1, ei, acc1, N_EDGES);
    combine_relu<64><<<(N_NODES * 64 + tb - 1) / tb, tb, 0, stream>>>(acc1, z1, cnt, N_NODES);

    // ---- layer 2 (64 -> 32): scatter only 32 channels ----
    zero_f32<<<2048, tb, 0, stream>>>(acc2, N_NODES * 32);
    gemm_dual_wmma<2><<<gemm_blocks, 256, 0, stream>>>(h1, W2l, W2r, b2l, y2, z2, N_NODES);
    scatter_add<32><<<(N_EDGES * 8 + tb - 1) / tb, tb, 0, stream>>>(y2, ei, acc2, N_EDGES);
    combine_relu<32><<<(N_NODES * 32 + tb - 1) / tb, tb, 0, stream>>>(acc2, z2, cnt, N_NODES);

    // ---- final 32 -> 1 ----
    final_linear<<<(N_NODES + tb - 1) / tb, tb, 0, stream>>>(h2, Wlin, blin, out, N_NODES);
}